// HSPPrefillCompressor_5944234738186
// MI455X (gfx1250) — compile-verified
//
#include <hip/hip_runtime.h>

// HSP prefill compressor for MI455X (gfx1250, wave32).
// One workgroup per batch row; 1024 threads (32 waves) x 16 contiguous tokens.
// Memory-bound (~48MB total traffic -> ~2us at 23.3 TB/s). No matmul exists in
// this op, so no WMMA; CDNA5 paths used: async global->LDS staging (ASYNCcnt),
// wave32 shuffle scans, LDS atomics, 320KB-WGP-LDS row residency.

#define B_     128
#define L_     16384
#define PAGE_  64
#define P_     (L_ / PAGE_)        // 256
#define BLOCK_ 1024
#define TPT_   (L_ / BLOCK_)       // 16 tokens per thread
#define NWAVE_ (BLOCK_ / 32)       // 32 waves
#define BND_   13
#define SEGW_  ((L_ + 1 + 31) / 32) // 513 dwords of segment bits

__global__ __launch_bounds__(BLOCK_) void hsp_compress_kernel(
    const int* __restrict__ input_ids,
    const int* __restrict__ attn_mask,
    const int* __restrict__ token2page,
    const int* __restrict__ keep_pages,
    const int* __restrict__ query_pos,
    int* __restrict__ out_ids,
    int* __restrict__ out_attn,
    int* __restrict__ out_pos)
{
    __shared__ int      ids_lds[L_];      // 64KB: whole row of input_ids
    __shared__ int      pk_lds[P_];       // 1KB: keep_pages row
    __shared__ unsigned segbits[SEGW_];   // sentence-segment "any kept" bitset
    __shared__ int      wsum[NWAVE_];     // wave scan totals
    __shared__ int      s_total;
    __shared__ int      s_lv;

    const int b    = blockIdx.x;
    const int t    = threadIdx.x;
    const int lane = t & 31;
    const int wid  = t >> 5;
    const int base = t * TPT_;

    const int* ids_row = input_ids  + (size_t)b * L_;
    const int* am_row  = attn_mask  + (size_t)b * L_;
    const int* t2p_row = token2page + (size_t)b * L_;
    const int* kp_row  = keep_pages + (size_t)b * P_;

    // ---- Phase 0: async-stage input_ids row into LDS (CDNA5 ASYNCcnt path) ----
    {
        unsigned lds0 = (unsigned)(size_t)(&ids_lds[0]);
        unsigned off  = (unsigned)(base * 4);
        #pragma unroll
        for (int j = 0; j < (TPT_ * 4) / 16; ++j) {
            unsigned o = off + (unsigned)(j * 16);
            asm volatile("global_load_async_to_lds_b128 %0, %1, %2"
                         :: "v"(lds0 + o), "v"(o), "s"(ids_row)
                         : "memory");
        }
    }
    if (t < P_) pk_lds[t] = kp_row[t];
    for (int i = t; i < SEGW_; i += BLOCK_) segbits[i] = 0u;
    if (t == 0) s_lv = -1;

    int qp = query_pos[b];
    qp = qp < 0 ? 0 : (qp > L_ - 1 ? L_ - 1 : qp);

    // vector loads of attention_mask / token2page for this thread's chunk
    int am[TPT_], tp[TPT_];
    {
        const int4* a4 = (const int4*)(am_row + base);
        const int4* p4 = (const int4*)(t2p_row + base);
        #pragma unroll
        for (int j = 0; j < TPT_ / 4; ++j) {
            int4 a = a4[j], p = p4[j];
            am[4*j+0] = a.x; am[4*j+1] = a.y; am[4*j+2] = a.z; am[4*j+3] = a.w;
            tp[4*j+0] = p.x; tp[4*j+1] = p.y; tp[4*j+2] = p.z; tp[4*j+3] = p.w;
        }
    }

    asm volatile("s_wait_asynccnt 0x0" ::: "memory");
    __syncthreads();

    // ---- Phase 1: per-token predicate bits packed into u32 bitfields ----
    unsigned validb = 0, maskb = 0, bcb = 0, keptb = 0;
    int lvloc = -1;
    #pragma unroll
    for (int i = 0; i < TPT_; ++i) {
        int p  = base + i;
        int id = ids_lds[p];
        bool valid  = am[i] != 0;
        int  pg     = tp[i]; pg = pg < 0 ? 0 : (pg > P_ - 1 ? P_ - 1 : pg);
        bool pk     = pk_lds[pg] != 0;
        bool in_ctx = p < qp;
        bool m0     = valid && pk;                 // page-level keep
        bool m      = m0 || (valid && !in_ctx);    // + keep from query_pos on
        bool bc     = valid && in_ctx && (id == BND_);
        bool kc     = m0 && in_ctx;                // kept_ctx for segment-OR
        if (valid) lvloc = p;
        validb |= ((unsigned)valid) << i;
        maskb  |= ((unsigned)m)     << i;
        bcb    |= ((unsigned)bc)    << i;
        keptb  |= ((unsigned)kc)    << i;
    }
    if (lvloc >= 0) atomicMax(&s_lv, lvloc);

    // ---- Phase 2: block exclusive scan of boundary counts -> segment base ----
    int bcnt = __popc(bcb);
    int incl = bcnt;
    #pragma unroll
    for (int off = 1; off < 32; off <<= 1) {
        int n = __shfl_up(incl, off, 32);
        if (lane >= off) incl += n;
    }
    if (lane == 31) wsum[wid] = incl;
    __syncthreads();
    if (wid == 0) {
        int s = wsum[lane];
        #pragma unroll
        for (int off = 1; off < 32; off <<= 1) {
            int n = __shfl_up(s, off, 32);
            if (lane >= off) s += n;
        }
        wsum[lane] = s;
    }
    __syncthreads();
    int segbase = (incl - bcnt) + (wid ? wsum[wid - 1] : 0);

    // ---- Phase 3: segment-OR via LDS bitset atomics ----
    #pragma unroll
    for (int i = 0; i < TPT_; ++i) {
        if (keptb & (1u << i)) {
            int seg = segbase + __popc(bcb & ((1u << i) - 1u));
            atomicOr(&segbits[seg >> 5], 1u << (seg & 31));
        }
    }
    __syncthreads();

    // ---- Phase 4: sentence-span merge ----
    unsigned finalb = maskb;
    #pragma unroll
    for (int i = 0; i < TPT_; ++i) {
        int p = base + i;
        if ((p < qp) && ((validb >> i) & 1u)) {
            int seg = segbase + __popc(bcb & ((1u << i) - 1u));
            if (segbits[seg >> 5] & (1u << (seg & 31))) finalb |= 1u << i;
        }
    }
    int kcnt = __popc(finalb);

    // ---- Phase 5: block scan of kept counts -> compaction offsets + count ----
    int incl2 = kcnt;
    #pragma unroll
    for (int off = 1; off < 32; off <<= 1) {
        int n = __shfl_up(incl2, off, 32);
        if (lane >= off) incl2 += n;
    }
    if (lane == 31) wsum[wid] = incl2;
    __syncthreads();
    if (wid == 0) {
        int s = wsum[lane];
        #pragma unroll
        for (int off = 1; off < 32; off <<= 1) {
            int n = __shfl_up(s, off, 32);
            if (lane >= off) s += n;
        }
        wsum[lane] = s;
        if (lane == 31) s_total = s;
    }
    __syncthreads();
    int obase = (incl2 - kcnt) + (wid ? wsum[wid - 1] : 0);
    int count = s_total;

    // ---- Fallback for empty rows: keep last valid token (else index 0) ----
    if (count == 0) {
        int fb = (s_lv >= 0) ? s_lv : 0;
        finalb = (fb >= base && fb < base + TPT_) ? (1u << (fb - base)) : 0u;
        obase  = 0;
        count  = 1;
    }

    // ---- Phase 6: outputs ----
    int* oid = out_ids  + (size_t)b * L_;
    int* oat = out_attn + (size_t)b * L_;
    int* opo = out_pos  + (size_t)b * L_;

    // attn / position ids / pad; vectorized. Positions >= count get id=0 here,
    // positions < count get exactly one scatter write below (disjoint sets).
    #pragma unroll
    for (int j = 0; j < TPT_ / 4; ++j) {
        int p0 = base + 4 * j;
        int4 a4i, p4i;
        a4i.x = (p0 + 0) < count; a4i.y = (p0 + 1) < count;
        a4i.z = (p0 + 2) < count; a4i.w = (p0 + 3) < count;
        p4i.x = a4i.x ? (p0 + 0) : 0; p4i.y = a4i.y ? (p0 + 1) : 0;
        p4i.z = a4i.z ? (p0 + 2) : 0; p4i.w = a4i.w ? (p0 + 3) : 0;
        *(int4*)(oat + p0) = a4i;
        *(int4*)(opo + p0) = p4i;
        #pragma unroll
        for (int k = 0; k < 4; ++k)
            if (p0 + k >= count) oid[p0 + k] = 0;  // PAD_ID
    }
    // stable compaction scatter of kept ids (read staged ids from LDS)
    int o = obase;
    #pragma unroll
    for (int i = 0; i < TPT_; ++i) {
        if (finalb & (1u << i)) {
            oid[o++] = ids_lds[base + i];
        }
    }
}

extern "C" void kernel_launch(void* const* d_in, const int* in_sizes, int n_in,
                              void* d_out, int out_size, void* d_ws, size_t ws_size,
                              hipStream_t stream) {
    (void)in_sizes; (void)n_in; (void)out_size; (void)d_ws; (void)ws_size;
    const int* input_ids  = (const int*)d_in[0];
    const int* attn_mask  = (const int*)d_in[1];
    // d_in[2] = token_valid: unused (attention_mask overrides it in reference)
    const int* token2page = (const int*)d_in[3];
    const int* keep_pages = (const int*)d_in[4];
    const int* query_pos  = (const int*)d_in[5];

    int* out = (int*)d_out;
    int* out_ids  = out;
    int* out_attn = out + (size_t)B_ * L_;
    int* out_pos  = out + 2 * (size_t)B_ * L_;

    hsp_compress_kernel<<<B_, BLOCK_, 0, stream>>>(
        input_ids, attn_mask, token2page, keep_pages, query_pos,
        out_ids, out_attn, out_pos);
}